// CentroidDistance_54296976556720
// MI455X (gfx1250) — compile-verified
//
#include <hip/hip_runtime.h>
#include <hip/hip_bf16.h>
#include <math.h>

typedef __attribute__((ext_vector_type(2))) float v2f;
typedef __attribute__((ext_vector_type(8))) float v8f;

#define NODE_NUM     65536
#define NUM_CENTROID 1024
#define DIM          64
#define EPS          1e-6f

// ---------------------------------------------------------------------------
// Kernel 1: c' = hyp_linear(expmap0(proj_tan0(centroid_weight)))  with the
// time component (col 0) pre-negated so the main GEMM is a plain dot product.
// One 64-thread block per centroid row.
// ---------------------------------------------------------------------------
__global__ void __launch_bounds__(DIM)
centroid_transform(const float* __restrict__ cw,
                   const float* __restrict__ W,
                   const float* __restrict__ b,
                   float* __restrict__ cprime) {
    __shared__ float s_x[DIM];
    __shared__ float s_red[DIM];
    const int i = blockIdx.x;   // centroid index
    const int j = threadIdx.x;  // feature index

    // proj_tan0: zero the time component
    float u = (j == 0) ? 0.0f : cw[i * DIM + j];

    // ||sp||
    s_red[j] = u * u;
    __syncthreads();
    if (j == 0) {
        float n2 = 0.0f;
        for (int k = 1; k < DIM; ++k) n2 += s_red[k];
        s_red[0] = sqrtf(fmaxf(n2, EPS));
    }
    __syncthreads();
    const float n = s_red[0];

    // expmap0
    const float x = (j == 0) ? coshf(n) : (sinhf(n) / n) * u;
    s_x[j] = x;
    __syncthreads();

    // y = x @ W.T + b   (row j of W is contiguous)
    float y = b[j];
    #pragma unroll 8
    for (int k = 0; k < DIM; ++k) y += s_x[k] * W[j * DIM + k];
    __syncthreads();  // protect s_red reuse

    // t = sqrt(1 + ||y[1:]||^2)
    s_red[j] = (j == 0) ? 0.0f : y * y;
    __syncthreads();
    if (j == 0) {
        float sy = 0.0f;
        for (int k = 1; k < DIM; ++k) sy += s_red[k];
        s_red[0] = sqrtf(1.0f + sy);
    }
    __syncthreads();
    const float t = s_red[0];

    // store with col 0 negated:  inner = node @ c'.T  directly
    cprime[i * DIM + j] = (j == 0) ? -t : y;
}

// ---------------------------------------------------------------------------
// Kernel 2: inner = node @ c'.T  via V_WMMA_F32_16X16X4_F32, fused with
// dist = arccosh(max(-inner, 1+eps)) * mask.
// Wave tile: 16x16.  Block tile: 32(M) x 64(N) = 8 waves.
// Output stores are non-temporal: 268 MB write-once stream must not evict
// the L2-resident A/B working set (node_repr 16 MB + c' 256 KB).
// ---------------------------------------------------------------------------
__global__ void __launch_bounds__(256)
centroid_dist_gemm(const float* __restrict__ node,
                   const float* __restrict__ mask,
                   const float* __restrict__ cp,
                   float* __restrict__ out) {
    const int lane = threadIdx.x & 31;
    const int wave = threadIdx.x >> 5;      // 0..7
    const int wm   = wave & 1;              // 0..1  -> M
    const int wn   = wave >> 1;             // 0..3  -> N

    const int m_base = blockIdx.y * 32 + wm * 16;
    const int n_base = blockIdx.x * 64 + wn * 16;

    // 32-bit A 16x4 layout: lanes 0-15 hold K = k0+{0,1}, lanes 16-31 K = k0+{2,3}
    const int lrow  = lane & 15;
    const int khalf = (lane >> 4) * 2;      // 0 or 2

    const float* aptr = node + (size_t)(m_base + lrow) * DIM + khalf;
    const float* bptr = cp   + (size_t)(n_base + lrow) * DIM + khalf;

    v8f acc = {};
    #pragma unroll
    for (int k0 = 0; k0 < DIM; k0 += 4) {
        v2f a = *(const v2f*)(aptr + k0);   // 8B-aligned float2
        v2f bb = *(const v2f*)(bptr + k0);
        // (neg_a, A, neg_b, B, c_mod, C, reuse_a, reuse_b)
        acc = __builtin_amdgcn_wmma_f32_16x16x4_f32(
            false, a, false, bb, (short)0, acc, false, false);
    }

    // C/D layout: VGPR r -> row (r + 8*(lane>=16)), col (lane & 15)
    const int col  = n_base + (lane & 15);
    const int rofs = (lane >> 4) * 8;
    #pragma unroll
    for (int r = 0; r < 8; ++r) {
        const int row = m_base + r + rofs;
        float v = -acc[r];
        v = fmaxf(v, 1.0f + EPS);
        const float d = logf(v + sqrtf(v * v - 1.0f));   // arccosh
        __builtin_nontemporal_store(d * mask[row],
                                    out + (size_t)row * NUM_CENTROID + col);
    }
}

// ---------------------------------------------------------------------------
extern "C" void kernel_launch(void* const* d_in, const int* in_sizes, int n_in,
                              void* d_out, int out_size, void* d_ws, size_t ws_size,
                              hipStream_t stream) {
    const float* node = (const float*)d_in[0];  // (65536, 64)
    const float* mask = (const float*)d_in[1];  // (65536, 1)
    const float* cw   = (const float*)d_in[2];  // (1024, 64)
    const float* W    = (const float*)d_in[3];  // (64, 64)
    const float* b    = (const float*)d_in[4];  // (64,)
    float* out    = (float*)d_out;              // (65536, 1024)
    float* cprime = (float*)d_ws;               // 1024*64*4 = 256 KB scratch

    centroid_transform<<<NUM_CENTROID, DIM, 0, stream>>>(cw, W, b, cprime);

    dim3 grid(NUM_CENTROID / 64, NODE_NUM / 32);  // (16, 2048)
    centroid_dist_gemm<<<grid, 256, 0, stream>>>(node, mask, cprime, out);
}